// GraphLayerBase_88596585382214
// MI455X (gfx1250) — compile-verified
//
#include <hip/hip_runtime.h>

// ---------------------------------------------------------------------------
// GraphLayer on MI455X (gfx1250, wave32, WMMA).
//
// Reference decomposition:
//   h3  = X @ g3_w^T + g3_b                 [N,128]  (WMMA f32 GEMM)
//   m2  = X @ g2_w^T + g2_b                 [N,128]  (WMMA f32 GEMM)
//   idx = top-3 nearest (L1) centers        [N,3]    (96KB LDS-resident scan)
//   w_k = dot(h3_i, h3_{idx_k})             (NOT the full N x N GEMM!)
//   msg = sum_k w_k * m2[idx_k] / 3         [N,128]
//   out = msg @ g5_w^T + g5_b + X           [N,128]  (WMMA f32 GEMM)
// ---------------------------------------------------------------------------

typedef __attribute__((ext_vector_type(2))) float v2f;
typedef __attribute__((ext_vector_type(4))) float v4f;
typedef __attribute__((ext_vector_type(8))) float v8f;

#define NROWS 8192
#define FDIM  128

// ---------------------------------------------------------------------------
// Kernel 1/4: Y = X @ W^T + bias (+ residual), W row-major [128,128].
// One wave -> 16 rows x all 128 cols. 8 x v8f accumulators.
// A frag (16x4 f32):  lane l<16: M=l,   K={k0,k0+1}; lane l>=16: M=l-16, K={k0+2,k0+3}
// B frag (4x16 f32):  lane l<16: N=l,   K={k0,k0+1}; lane l>=16: N=l-16, K={k0+2,k0+3}
//   B[k][n] = W[n][k] -> b64 load at W + n*128 + k0 + 2*(lane>=16)
// ---------------------------------------------------------------------------
__global__ __launch_bounds__(256) void gemm_nt_bias_kernel(
    const float* __restrict__ X,
    const float* __restrict__ W,
    const float* __restrict__ bias,
    const float* __restrict__ residual,   // may be nullptr
    float* __restrict__ Y)
{
    const int wave    = threadIdx.x >> 5;          // 0..7
    const int lane    = threadIdx.x & 31;
    const int l       = lane & 15;
    const int lhalf   = lane >> 4;                 // 0 or 1
    const int rowBase = (blockIdx.x * 8 + wave) * 16;

    v8f acc[8];
    const v8f zero = {0.f, 0.f, 0.f, 0.f, 0.f, 0.f, 0.f, 0.f};
#pragma unroll
    for (int t = 0; t < 8; ++t) acc[t] = zero;

    const float* xrow = X + (size_t)(rowBase + l) * FDIM + 2 * lhalf;

    for (int k = 0; k < FDIM; k += 4) {
        v2f a = *(const v2f*)(xrow + k);
#pragma unroll
        for (int t = 0; t < 8; ++t) {
            const int n = t * 16 + l;
            v2f b = *(const v2f*)(W + (size_t)n * FDIM + k + 2 * lhalf);
            acc[t] = __builtin_amdgcn_wmma_f32_16x16x4_f32(
                /*neg_a=*/false, a, /*neg_b=*/false, b,
                /*c_mod=*/(short)0, acc[t],
                /*reuse_a=*/false, /*reuse_b=*/false);
        }
    }

    // D layout: VGPR r, lane<16 -> (M=r, N=l); lane>=16 -> (M=8+r, N=l)
#pragma unroll
    for (int t = 0; t < 8; ++t) {
        const int col = t * 16 + l;
        const float bv = bias[col];
#pragma unroll
        for (int r = 0; r < 8; ++r) {
            const int row = rowBase + r + 8 * lhalf;
            float v = acc[t][r] + bv;
            if (residual) v += residual[(size_t)row * FDIM + col];
            Y[(size_t)row * FDIM + col] = v;
        }
    }
}

// ---------------------------------------------------------------------------
// Kernel 2/4: top-3 nearest neighbors by L1 distance of centers.
// centers[j] = st[j,0:3] + st[j,3:6].  All 8192 centers = 96 KB -> LDS.
// Strict '<' + ascending scan reproduces stable-argsort tie-breaking.
// ---------------------------------------------------------------------------
__global__ __launch_bounds__(256) void topk3_kernel(
    const float* __restrict__ st,
    int* __restrict__ idxs)
{
    __shared__ float cx[NROWS];
    __shared__ float cy[NROWS];
    __shared__ float cz[NROWS];

    for (int j = threadIdx.x; j < NROWS; j += 256) {
        cx[j] = st[j * 6 + 0] + st[j * 6 + 3];
        cy[j] = st[j * 6 + 1] + st[j * 6 + 4];
        cz[j] = st[j * 6 + 2] + st[j * 6 + 5];
    }
    __syncthreads();

    const int i = blockIdx.x * 256 + threadIdx.x;
    const float x = cx[i], y = cy[i], z = cz[i];

    float d0 = __builtin_inff(), d1 = d0, d2 = d0;
    int   i0 = 0, i1 = 0, i2 = 0;

    for (int j = 0; j < NROWS; ++j) {
        const float d = __builtin_fabsf(cx[j] - x)
                      + __builtin_fabsf(cy[j] - y)
                      + __builtin_fabsf(cz[j] - z);
        if (j == i) continue;
        if (d < d0)      { d2 = d1; i2 = i1; d1 = d0; i1 = i0; d0 = d; i0 = j; }
        else if (d < d1) { d2 = d1; i2 = i1; d1 = d;  i1 = j; }
        else if (d < d2) { d2 = d;  i2 = j; }
    }
    idxs[i * 3 + 0] = i0;
    idxs[i * 3 + 1] = i1;
    idxs[i * 3 + 2] = i2;
}

// ---------------------------------------------------------------------------
// Kernel 3/4: per-row message. One wave (32 lanes) per row i.
//   w_k = dot(h3[i], h3[idx_k])   (length-128 dot, v4f loads, shfl butterfly)
//   msg[i] = (w0*m2[j0] + w1*m2[j1] + w2*m2[j2]) / 3
// ---------------------------------------------------------------------------
__global__ __launch_bounds__(256) void message_kernel(
    const float* __restrict__ h3,
    const float* __restrict__ m2,
    const int*   __restrict__ idxs,
    float* __restrict__ msg)
{
    const int wave = threadIdx.x >> 5;
    const int lane = threadIdx.x & 31;
    const int i    = blockIdx.x * 8 + wave;

    const int j0 = idxs[i * 3 + 0];
    const int j1 = idxs[i * 3 + 1];
    const int j2 = idxs[i * 3 + 2];

    const v4f a  = *(const v4f*)(h3 + (size_t)i  * FDIM + lane * 4);
    const v4f b0 = *(const v4f*)(h3 + (size_t)j0 * FDIM + lane * 4);
    const v4f b1 = *(const v4f*)(h3 + (size_t)j1 * FDIM + lane * 4);
    const v4f b2 = *(const v4f*)(h3 + (size_t)j2 * FDIM + lane * 4);

    float p0 = a.x * b0.x + a.y * b0.y + a.z * b0.z + a.w * b0.w;
    float p1 = a.x * b1.x + a.y * b1.y + a.z * b1.z + a.w * b1.w;
    float p2 = a.x * b2.x + a.y * b2.y + a.z * b2.z + a.w * b2.w;

#pragma unroll
    for (int off = 16; off > 0; off >>= 1) {   // wave32 butterfly: all lanes get sum
        p0 += __shfl_xor(p0, off, 32);
        p1 += __shfl_xor(p1, off, 32);
        p2 += __shfl_xor(p2, off, 32);
    }

    const v4f m0 = *(const v4f*)(m2 + (size_t)j0 * FDIM + lane * 4);
    const v4f m1 = *(const v4f*)(m2 + (size_t)j1 * FDIM + lane * 4);
    const v4f mm2 = *(const v4f*)(m2 + (size_t)j2 * FDIM + lane * 4);

    const float inv3 = 1.0f / 3.0f;
    v4f s;
    s.x = (p0 * m0.x + p1 * m1.x + p2 * mm2.x) * inv3;
    s.y = (p0 * m0.y + p1 * m1.y + p2 * mm2.y) * inv3;
    s.z = (p0 * m0.z + p1 * m1.z + p2 * mm2.z) * inv3;
    s.w = (p0 * m0.w + p1 * m1.w + p2 * mm2.w) * inv3;

    *(v4f*)(msg + (size_t)i * FDIM + lane * 4) = s;
}

// ---------------------------------------------------------------------------
extern "C" void kernel_launch(void* const* d_in, const int* in_sizes, int n_in,
                              void* d_out, int out_size, void* d_ws, size_t ws_size,
                              hipStream_t stream) {
    const float* nodes = (const float*)d_in[0];   // [N,128]
    const float* st    = (const float*)d_in[1];   // [N,6]
    const float* g2w   = (const float*)d_in[2];
    const float* g2b   = (const float*)d_in[3];
    const float* g3w   = (const float*)d_in[4];
    const float* g3b   = (const float*)d_in[5];
    const float* g5w   = (const float*)d_in[6];
    const float* g5b   = (const float*)d_in[7];

    float* h3  = (float*)d_ws;                    // [N,128]
    float* m2  = h3 + (size_t)NROWS * FDIM;       // [N,128]
    float* msg = m2 + (size_t)NROWS * FDIM;       // [N,128]
    int*   idx = (int*)(msg + (size_t)NROWS * FDIM); // [N,3]

    const int gemmBlocks = NROWS / (16 * 8);      // 64

    gemm_nt_bias_kernel<<<gemmBlocks, 256, 0, stream>>>(nodes, g3w, g3b, nullptr, h3);
    gemm_nt_bias_kernel<<<gemmBlocks, 256, 0, stream>>>(nodes, g2w, g2b, nullptr, m2);
    topk3_kernel<<<NROWS / 256, 256, 0, stream>>>(st, idx);
    message_kernel<<<NROWS / 8, 256, 0, stream>>>(h3, m2, idx, msg);
    gemm_nt_bias_kernel<<<gemmBlocks, 256, 0, stream>>>(msg, g5w, g5b, nodes, (float*)d_out);
}